// DesNet_87540023427465
// MI455X (gfx1250) — compile-verified
//
#include <hip/hip_runtime.h>

#define NN 4096
#define H 5
#define ROWSPLIT 16
#define TPB 256   // 8 waves of 32 on gfx1250

typedef __attribute__((ext_vector_type(2))) float v2f;
typedef __attribute__((ext_vector_type(8))) float v8f;

__device__ __forceinline__ float tiny_mlp(float x,
    const float w1[H], const float b1[H], const float w2[H], float b2) {
  float acc = b2;
#pragma unroll
  for (int i = 0; i < H; ++i) {
    float h = fmaxf(fmaf(x, w1[i], b1[i]), 0.0f);   // relu(Linear(1,H))
    acc = fmaf(h, w2[i], acc);                      // Linear(H,1)
  }
  return fmaxf(acc, 0.0f);                          // final relu
}

__global__ void zero_out_kernel(float* __restrict__ out) {
  int i = blockIdx.x * blockDim.x + threadIdx.x;
  if (i < NN) out[i] = 0.0f;
}

// Zm[j] = mlpZ(Zj[j]) precomputed once (4096 elements, trivial cost).
__global__ void zmlp_kernel(const float* __restrict__ Zj, float* __restrict__ Zm,
                            const float* __restrict__ zw1, const float* __restrict__ zb1,
                            const float* __restrict__ zw2, const float* __restrict__ zb2) {
  float w1[H], b1[H], w2[H];
#pragma unroll
  for (int i = 0; i < H; ++i) { w1[i] = zw1[i]; b1[i] = zb1[i]; w2[i] = zw2[i]; }
  float b2 = zb2[0];
  int j = blockIdx.x * blockDim.x + threadIdx.x;
  if (j < NN) Zm[j] = tiny_mlp(Zj[j], w1, b1, w2, b2);
}

// out[c0+n] += sum_j Zm[j] * mlpR(R[j, c0+n]) via V_WMMA_F32_16X16X4_F32.
// A[m][k] = Zm[j0+k] (replicated over M); B[k][n] = mlpR(R[j0+k, c0+n]).
// f32 A 16x4 layout: lanes 0-15 -> {V0:K0, V1:K1}; lanes 16-31 -> {V0:K2, V1:K3}.
// f32 B 4x16 uses the same K striping (lane-half picks K pair, VGPR picks within),
// so pairing a.{x,y} with b.{x,y} from rows (j0+2*half, j0+2*half+1) is consistent.
__global__ void __launch_bounds__(TPB) zr_gemv_kernel(
    const float* __restrict__ R, const float* __restrict__ Zm,
    const float* __restrict__ rw1p, const float* __restrict__ rb1p,
    const float* __restrict__ rw2p, const float* __restrict__ rb2p,
    float* __restrict__ out) {
  float w1[H], b1[H], w2[H];
#pragma unroll
  for (int i = 0; i < H; ++i) { w1[i] = rw1p[i]; b1[i] = rb1p[i]; w2[i] = rw2p[i]; }
  const float b2 = rb2p[0];

  const int c0   = blockIdx.x * 16;          // 16-column output tile
  const int wave = threadIdx.x >> 5;
  const int lane = threadIdx.x & 31;
  const int ln   = lane & 15;                // N index within tile
  const int kh   = lane >> 4;                // K half (0: K0/K1, 1: K2/K3)

  const int rows_per_block = NN / ROWSPLIT;              // 256
  const int rows_per_wave  = rows_per_block / (TPB/32);  // 32
  const int jw = blockIdx.y * rows_per_block + wave * rows_per_wave;

  v8f acc = {};
  const float* colp = R + (size_t)c0 + (size_t)ln;

#pragma unroll
  for (int it = 0; it < rows_per_wave / 4; ++it) {       // 8 WMMAs per wave
    const int j0 = jw + it * 4;
    const int r0 = j0 + kh * 2;
    // Coalesced: lanes 0-15 read 64B of row r0 / r0+1; lanes 16-31 rows r0+2/r0+3.
    float x0 = colp[(size_t)r0 * NN];
    float x1 = colp[(size_t)(r0 + 1) * NN];
    v2f a, b;
    a.x = Zm[r0];
    a.y = Zm[r0 + 1];
    b.x = tiny_mlp(x0, w1, b1, w2, b2);
    b.y = tiny_mlp(x1, w1, b1, w2, b2);
    // 8 args: (neg_a, A, neg_b, B, c_mod, C, reuse_a, reuse_b)
    acc = __builtin_amdgcn_wmma_f32_16x16x4_f32(
        false, a, false, b, (short)0, acc, false, false);
  }

  // D[0][n] (VGPR0, lanes 0-15) holds the 16 column partials.
  __shared__ float red[TPB / 32][16];
  if (lane < 16) red[wave][ln] = acc[0];
  __syncthreads();
  if (wave == 0 && lane < 16) {
    float s = 0.0f;
#pragma unroll
    for (int r = 0; r < TPB / 32; ++r) s += red[r][ln];
    unsafeAtomicAdd(&out[c0 + ln], s);   // global_atomic_add_f32
  }
}

extern "C" void kernel_launch(void* const* d_in, const int* in_sizes, int n_in,
                              void* d_out, int out_size, void* d_ws, size_t ws_size,
                              hipStream_t stream) {
  const float* Rij = (const float*)d_in[0];
  const float* Zj  = (const float*)d_in[1];
  const float* rw1 = (const float*)d_in[2];
  const float* rb1 = (const float*)d_in[3];
  const float* rw2 = (const float*)d_in[4];
  const float* rb2 = (const float*)d_in[5];
  const float* zw1 = (const float*)d_in[6];
  const float* zb1 = (const float*)d_in[7];
  const float* zw2 = (const float*)d_in[8];
  const float* zb2 = (const float*)d_in[9];
  float* out = (float*)d_out;
  float* Zm  = (float*)d_ws;   // 4096 floats of scratch

  zero_out_kernel<<<(NN + 255) / 256, 256, 0, stream>>>(out);
  zmlp_kernel<<<(NN + 255) / 256, 256, 0, stream>>>(Zj, Zm, zw1, zb1, zw2, zb2);

  dim3 grid(NN / 16, ROWSPLIT);   // 256 column tiles x 16 row splits = 4096 blocks
  zr_gemv_kernel<<<grid, TPB, 0, stream>>>(Rij, Zm, rw1, rb1, rw2, rb2, out);
}